// ScaledMinPlus_70832600646265
// MI455X (gfx1250) — compile-verified
//
#include <hip/hip_runtime.h>
#include <math.h>

// Problem constants (from reference): B=2048, O=512, I=256, all fp32.
#define DIM_B 2048
#define DIM_O 512
#define DIM_I 256

#define BM 64          // rows (b) per block
#define BN 64          // cols (o) per block
#define LDK 260        // 256 + 4 dword pad: 1040B row stride, 16B aligned, rotates banks by 4/row

// ---- CDNA5 async global->LDS copy path (gfx1250) --------------------------
#if defined(__has_builtin)
#if __has_builtin(__builtin_amdgcn_global_load_async_to_lds_b128) && \
    __has_builtin(__builtin_amdgcn_s_wait_asynccnt)
#define USE_ASYNC_LDS 1
#endif
#if __has_builtin(__builtin_amdgcn_sched_barrier)
#define SCHED_FENCE() __builtin_amdgcn_sched_barrier(0)
#endif
#endif
#ifndef SCHED_FENCE
#define SCHED_FENCE() asm volatile("" ::: "memory")
#endif

typedef int v4i __attribute__((ext_vector_type(4)));
typedef __attribute__((address_space(1))) v4i* gv4i_p;   // global (AS1) int4*
typedef __attribute__((address_space(3))) v4i* lv4i_p;   // LDS    (AS3) int4*

__device__ __forceinline__ void copy_b128_to_lds(const float* __restrict__ gsrc,
                                                 float* __restrict__ ldst) {
#ifdef USE_ASYNC_LDS
    __builtin_amdgcn_global_load_async_to_lds_b128(
        (gv4i_p)(v4i*)(void*)gsrc,
        (lv4i_p)(v4i*)(void*)ldst,
        /*imm offset=*/0, /*cpol=*/0);
#else
    *(float4*)ldst = *(const float4*)gsrc;
#endif
}

// out[b,o] = min_i (W[o,i] + X[b,i] * F[o])
// launch_bounds(256, 4): min 4 waves/SIMD => <=256 VGPRs/wave. This matters:
// with a looser bound the compiler allocates VGPRs >255 and drowns the kernel
// in s_set_vgpr_msb traffic + spills.
__global__ __launch_bounds__(256, 4)
void ScaledMinPlus_70832600646265_kernel(const float* __restrict__ X,
                                         const float* __restrict__ W,
                                         const float* __restrict__ F,
                                         float* __restrict__ out) {
    __shared__ __align__(16) float Xs[BM * LDK];   // 66,560 B
    __shared__ __align__(16) float Ws[BN * LDK];   // 66,560 B  (total 130 KB < 320 KB WGP LDS)

    const int tid     = threadIdx.x;          // 0..255 (8 waves of 32)
    const int rowTile = blockIdx.y * BM;      // b-tile origin
    const int colTile = blockIdx.x * BN;      // o-tile origin

    // ---- Stage full-K tiles into LDS (async DMA-style copies on CDNA5) ----
    // 64 rows x 64 float4 chunks per matrix = 4096 chunks; 16 b128 copies/thread/matrix.
    #pragma unroll
    for (int it = 0; it < (BM * (DIM_I / 4)) / 256; ++it) {
        const int t   = tid + it * 256;
        const int row = t >> 6;               // chunk-row
        const int c4  = (t & 63) << 2;        // dword offset in row
        copy_b128_to_lds(&X[(size_t)(rowTile + row) * DIM_I + c4], &Xs[row * LDK + c4]);
    }
    #pragma unroll
    for (int it = 0; it < (BN * (DIM_I / 4)) / 256; ++it) {
        const int t   = tid + it * 256;
        const int row = t >> 6;
        const int c4  = (t & 63) << 2;
        copy_b128_to_lds(&W[(size_t)(colTile + row) * DIM_I + c4], &Ws[row * LDK + c4]);
    }
#ifdef USE_ASYNC_LDS
    __builtin_amdgcn_s_wait_asynccnt(0);      // this wave's async copies landed in LDS
#endif
    __syncthreads();                          // all waves' copies visible

    // ---- 4x4 register micro-tile per thread ----
    // rows: b = rowTile + ty + 16*ri ; cols: o = colTile + tx + 16*ci
    // (16-strided cols => the 16 distinct W b128 reads/wave land on banks
    //  4*tx..4*tx+3: all 64 banks exactly once, conflict-free)
    const int tx = tid & 15;
    const int ty = tid >> 4;

    // Base pointers: all inner-loop DS accesses become base + immediate offset
    // (max offset 48*LDK*4 + 1008 = 50928 B, fits the 16-bit DS offset field).
    const float* __restrict__ xb = &Xs[ty * LDK];
    const float* __restrict__ wb = &Ws[tx * LDK];

    float f[4];
    #pragma unroll
    for (int ci = 0; ci < 4; ++ci)
        f[ci] = F[colTile + tx + 16 * ci];

    float acc[4][4];
    #pragma unroll
    for (int ri = 0; ri < 4; ++ri)
        #pragma unroll
        for (int ci = 0; ci < 4; ++ci)
            acc[ri][ci] = INFINITY;

    #pragma unroll 2
    for (int k = 0; k < DIM_I; k += 4) {
        float4 xv[4], wv[4];
        #pragma unroll
        for (int ri = 0; ri < 4; ++ri)
            xv[ri] = *(const float4*)(xb + (16 * ri) * LDK + k);
        #pragma unroll
        for (int ci = 0; ci < 4; ++ci)
            wv[ci] = *(const float4*)(wb + (16 * ci) * LDK + k);

        #pragma unroll
        for (int ri = 0; ri < 4; ++ri) {
            #pragma unroll
            for (int ci = 0; ci < 4; ++ci) {
                const float m0 = fminf(__fmaf_rn(xv[ri].x, f[ci], wv[ci].x),
                                       __fmaf_rn(xv[ri].y, f[ci], wv[ci].y));
                const float m1 = fminf(__fmaf_rn(xv[ri].z, f[ci], wv[ci].z),
                                       __fmaf_rn(xv[ri].w, f[ci], wv[ci].w));
                acc[ri][ci] = fminf(acc[ri][ci], fminf(m0, m1));  // -> v_min3_num_f32
            }
        }
        SCHED_FENCE();   // keep the scheduler from hoisting all DS loads at once
    }

    // ---- Writeback: lanes 0..15 hit 16 consecutive o's -> coalesced ----
    #pragma unroll
    for (int ri = 0; ri < 4; ++ri) {
        const size_t b = (size_t)(rowTile + ty + 16 * ri);
        #pragma unroll
        for (int ci = 0; ci < 4; ++ci)
            out[b * DIM_O + colTile + tx + 16 * ci] = acc[ri][ci];
    }
}

extern "C" void kernel_launch(void* const* d_in, const int* in_sizes, int n_in,
                              void* d_out, int out_size, void* d_ws, size_t ws_size,
                              hipStream_t stream) {
    (void)in_sizes; (void)n_in; (void)out_size; (void)d_ws; (void)ws_size;
    const float* X = (const float*)d_in[0];   // (B, I)
    const float* W = (const float*)d_in[1];   // (O, I)
    const float* F = (const float*)d_in[2];   // (O, 1)
    float*     out = (float*)d_out;           // (B, O)

    dim3 grid(DIM_O / BN, DIM_B / BM);        // (8, 32) = 256 workgroups
    dim3 block(256);                          // 8 wave32's
    ScaledMinPlus_70832600646265_kernel<<<grid, block, 0, stream>>>(X, W, F, out);
}